// SurfaceLoss_3204045603274
// MI455X (gfx1250) — compile-verified
//
#include <hip/hip_runtime.h>
#include <hip/hip_bf16.h>
#include <math.h>

typedef float v2f __attribute__((ext_vector_type(2)));
typedef float v8f __attribute__((ext_vector_type(8)));

#define BATCH 8
#define MKEY  1024
#define NPC   16384
#define WAVES_PER_BLOCK 4
#define TILE_M 16
#define TILE_N 16
#define NITER (NPC / (WAVES_PER_BLOCK * TILE_N))   // 256, compile-time trip count
#define STRIDE (WAVES_PER_BLOCK * TILE_N)          // 64 columns per block-iteration

// Raw v_min_num_f32 without fminf's NaN-canonicalization (inputs are NaN-free).
__device__ __forceinline__ float fmin_raw(float x, float y) {
    float r;
    asm("v_min_num_f32 %0, %1, %2" : "=v"(r) : "v"(x), "v"(y));
    return r;
}

// One block = one (b, 16-keypoint tile); 4 waves split the N range.
// K=4 pad-slot trick: A = [-2k0,-2k1 | -2k2, 1.0], B = [p0,p1 | p2, psq],
// C = inline 0  =>  D = psq - 2*(k.p) in a single v_wmma_f32_16x16x4_f32,
// no C broadcast movs. Raw v_min accumulate; shuffle+LDS min combine;
// sqrt deferred to one op per keypoint (monotonicity of sqrt over min).
__global__ void __launch_bounds__(WAVES_PER_BLOCK * 32)
surface_min_kernel(const float* __restrict__ keypoint,
                   const float* __restrict__ pc,
                   float* __restrict__ min_dist /* [BATCH*MKEY] */) {
    const int tile = blockIdx.x;          // 0..63  (M/16 tiles)
    const int b    = blockIdx.y;          // 0..7
    const int m_base = tile * TILE_M;

    const int tid  = threadIdx.x;
    const int wave = tid >> 5;            // wave32
    const int lane = tid & 31;
    const int half = lane >> 4;           // 0: K rows 0/1, 1: K rows 2/3
    const int l16  = lane & 15;

    const float* kp0 = keypoint + (size_t)b * 3 * MKEY;
    const float* pc0 = pc       + (size_t)b * 3 * NPC;

    // ---- A operand: 16x4 f32; K row 3 carries the psq pass-through (1.0) ----
    v2f a;
    {
        const int m = m_base + l16;
        if (half == 0) {
            a.x = -2.0f * kp0[0 * MKEY + m];
            a.y = -2.0f * kp0[1 * MKEY + m];
        } else {
            a.x = -2.0f * kp0[2 * MKEY + m];
            a.y = 1.0f;                    // multiplies B's psq row
        }
    }

    // Per-lane column base pointer: coord-0 row, column (wave*16 + l16).
    const float* col = pc0 + wave * TILE_N + l16;

    v8f mind;
    #pragma unroll
    for (int i = 0; i < 8; ++i) mind[i] = 3.0e38f;

    #pragma unroll 8
    for (int it = 0; it < NITER; ++it) {
        const int off = it * STRIDE;

        // speculative prefetch of next chunk's coord-0 line (stays in pc buffer)
        __builtin_prefetch(&col[off + STRIDE], 0, 0);

        const float p0 = col[off];
        const float p1 = col[off + NPC];
        const float p2 = col[off + 2 * NPC];
        const float psq = fmaf(p2, p2, fmaf(p1, p1, p0 * p0));

        // B operand: 4x16; K rows striped across lane halves, row 3 = psq.
        v2f bb;
        bb.x = (half == 0) ? p0 : p2;
        bb.y = (half == 0) ? p1 : psq;

        // D = A*B + 0 = psq - 2*dot   (256 pair evaluations per wmma)
        v8f c = {};
        v8f d = __builtin_amdgcn_wmma_f32_16x16x4_f32(
            false, a, false, bb, (short)0, c, false, false);

        #pragma unroll
        for (int i = 0; i < 8; ++i) mind[i] = fmin_raw(mind[i], d[i]);
    }

    // ---- reduce over columns: width-16 xor tree (halves hold distinct M) ----
    #pragma unroll
    for (int i = 0; i < 8; ++i) {
        #pragma unroll
        for (int m_ = 8; m_ >= 1; m_ >>= 1)
            mind[i] = fmin_raw(mind[i], __shfl_xor(mind[i], m_, 16));
    }

    // lane 0 holds min for M = 0..7 (VGPR v -> M=v); lane 16 for M = 8..15.
    __shared__ float smem[WAVES_PER_BLOCK][TILE_M];
    if (l16 == 0) {
        #pragma unroll
        for (int i = 0; i < 8; ++i) smem[wave][half * 8 + i] = mind[i];
    }
    __syncthreads();

    // ---- combine waves, add ksq back, sqrt, store per-keypoint distance ----
    if (tid < TILE_M) {
        float v = fmin_raw(fmin_raw(smem[0][tid], smem[1][tid]),
                           fmin_raw(smem[2][tid], smem[3][tid]));
        const int m = m_base + tid;
        const float k0 = kp0[0 * MKEY + m];
        const float k1 = kp0[1 * MKEY + m];
        const float k2 = kp0[2 * MKEY + m];
        const float ksq = fmaf(k2, k2, fmaf(k1, k1, k0 * k0));
        float d2 = fmaxf(v + ksq, 0.0f);
        min_dist[(size_t)b * MKEY + m] = sqrtf(d2);
    }
}

// Mean of BATCH*MKEY per-keypoint min distances -> scalar.
__global__ void __launch_bounds__(256)
mean_reduce_kernel(const float* __restrict__ part, float* __restrict__ out) {
    __shared__ float s[256];
    float acc = 0.0f;
    for (int i = threadIdx.x; i < BATCH * MKEY; i += 256) acc += part[i];
    s[threadIdx.x] = acc;
    __syncthreads();
    for (int off = 128; off > 0; off >>= 1) {
        if (threadIdx.x < off) s[threadIdx.x] += s[threadIdx.x + off];
        __syncthreads();
    }
    if (threadIdx.x == 0) out[0] = s[0] * (1.0f / (BATCH * MKEY));
}

extern "C" void kernel_launch(void* const* d_in, const int* in_sizes, int n_in,
                              void* d_out, int out_size, void* d_ws, size_t ws_size,
                              hipStream_t stream) {
    const float* keypoint = (const float*)d_in[0];   // 8 x 3 x 1024
    const float* pc       = (const float*)d_in[1];   // 8 x 3 x 16384
    float* out = (float*)d_out;                      // scalar
    float* ws  = (float*)d_ws;                       // >= 8192 floats

    dim3 grid(MKEY / TILE_M, BATCH);                 // 64 x 8 tiles
    dim3 block(WAVES_PER_BLOCK * 32);                // 128 threads = 4 waves
    surface_min_kernel<<<grid, block, 0, stream>>>(keypoint, pc, ws);
    mean_reduce_kernel<<<1, 256, 0, stream>>>(ws, out);
}